// FullyConnectedTP_18511309046149
// MI455X (gfx1250) — compile-verified
//
#include <hip/hip_runtime.h>

// CDNA5 / gfx1250. wave32. One edge per wave.
// out(16x4) = W_all^T(16x64) x S(64x4), done as 16 chained
// V_WMMA_F32_16X16X4_F32 ops (K=4 each, exact f32 math).

typedef __attribute__((ext_vector_type(2))) float v2f;
typedef __attribute__((ext_vector_type(4))) float v4f;
typedef __attribute__((ext_vector_type(8))) float v8f;

#define WAVES_PER_BLOCK 8

__global__ __launch_bounds__(32 * WAVES_PER_BLOCK)
void tp_fc_kernel(const float* __restrict__ u,
                  const float* __restrict__ v,
                  const float* __restrict__ w,
                  float* __restrict__ out,
                  int n_edges)
{
    const int lane    = threadIdx.x & 31;
    const int wave_id = threadIdx.x >> 5;
    const int e       = blockIdx.x * WAVES_PER_BLOCK + wave_id;

    // per-wave 64x4 f32 source matrix S (1 KB per wave)
    __shared__ float sS[WAVES_PER_BLOCK * 64 * 4];
    float* S = sS + wave_id * 256;

    if (e >= n_edges) return;

    // PW0 = sqrt(1/32); PW1*INV_SQRT3 == PW0 exactly; C4 = PW0/sqrt(3) = sqrt(1/96)
    const float PW0 = 0.1767766952966368811f;
    const float C4  = 0.1020620726159657527f;

    const float* up = u + (size_t)e * 64;
    const float* vp = v + (size_t)e * 4;
    const float v0  = vp[0];
    const float v1x = vp[1], v1y = vp[2], v1z = vp[3];

    // ---- Phase 1: each lane builds rows 2*lane and 2*lane+1 of S --------
#pragma unroll
    for (int t = 0; t < 2; ++t) {
        const int r  = 2 * lane + t;   // 0..63
        const int m  = r >> 4;         // which weight matrix block
        const int uu = r & 15;         // u-channel
        float a0 = 0.f, a1 = 0.f, a2 = 0.f, a3 = 0.f;
        if (m == 0) {
            const float x = up[uu];
            a0 = PW0 * v0 * x;
        } else if (m == 1) {
            const float x = up[uu];
            a1 = PW0 * x * v1x; a2 = PW0 * x * v1y; a3 = PW0 * x * v1z;
        } else if (m == 2) {
            const float x0 = up[16 + 3 * uu + 0];
            const float x1 = up[16 + 3 * uu + 1];
            const float x2 = up[16 + 3 * uu + 2];
            a1 = PW0 * v0 * x0; a2 = PW0 * v0 * x1; a3 = PW0 * v0 * x2;
        } else {
            const float x0 = up[16 + 3 * uu + 0];
            const float x1 = up[16 + 3 * uu + 1];
            const float x2 = up[16 + 3 * uu + 2];
            a0 = C4 * (x0 * v1x + x1 * v1y + x2 * v1z);
        }
        v4f row; row.x = a0; row.y = a1; row.z = a2; row.w = a3;
        *(v4f*)(S + r * 4) = row;   // ds_store_b128
    }

    // LDS pipe is in-order per wave; wait for our stores so cross-lane
    // reads below see the data. "memory" clobber stops compiler reordering.
    asm volatile("s_wait_dscnt 0" ::: "memory");

    // ---- Phase 2: 16 chained WMMAs, weights streamed from HBM into B ----
    const int  h      = lane >> 4;     // lane half: selects K = 2h + {0,1}
    const int  ko     = lane & 15;     // A: row kout; B: column w
    const bool act    = (ko < 4);
    const int  kosafe = ko & 3;

    // B[k = 2h+g, j = ko] = weight[e*1024 + (4c + k)*16 + j]
    const float* wp = w + (size_t)e * 1024 + h * 32 + ko;

    v8f acc = {};
#pragma unroll
    for (int c = 0; c < 16; ++c) {
        const int ra = 4 * c + 2 * h;          // contraction rows this lane feeds
        const float s0 = S[(ra + 0) * 4 + kosafe];
        const float s1 = S[(ra + 1) * 4 + kosafe];
        v2f A; A.x = act ? s0 : 0.f; A.y = act ? s1 : 0.f;
        v2f B; B.x = wp[64 * c]; B.y = wp[64 * c + 16];
        // D(16x16) += A(16x4) x B(4x16); rows 0..3 of D = out[kout][w]
        acc = __builtin_amdgcn_wmma_f32_16x16x4_f32(
            /*neg_a=*/false, A, /*neg_b=*/false, B,
            /*c_mod=*/(short)0, acc, /*reuse_a=*/false, /*reuse_b=*/false);
    }

    // ---- Store: lane L<16 holds D[0..3][w=L] in acc[0..3] ---------------
    if (lane < 16) {
        float* op = out + (size_t)e * 64;
        op[lane] = acc[0];                       // out0[w]
        float* o1 = op + 16 + 3 * lane;          // out1[w][0..2]
        o1[0] = acc[1]; o1[1] = acc[2]; o1[2] = acc[3];
    }
}

extern "C" void kernel_launch(void* const* d_in, const int* in_sizes, int n_in,
                              void* d_out, int out_size, void* d_ws, size_t ws_size,
                              hipStream_t stream)
{
    const float* u  = (const float*)d_in[0];   // (N, 64)
    const float* v  = (const float*)d_in[1];   // (N, 4)
    const float* w  = (const float*)d_in[2];   // (N, 1024)
    float* out      = (float*)d_out;           // (N, 64)

    const int n = in_sizes[0] / 64;
    const int blocks = (n + WAVES_PER_BLOCK - 1) / WAVES_PER_BLOCK;
    tp_fc_kernel<<<blocks, 32 * WAVES_PER_BLOCK, 0, stream>>>(u, v, w, out, n);
}